// HybridRUCSupervised_67327907332624
// MI455X (gfx1250) — compile-verified
//
#include <hip/hip_runtime.h>

typedef __attribute__((ext_vector_type(2))) float v2f;
typedef __attribute__((ext_vector_type(4))) float v4f;
typedef __attribute__((ext_vector_type(8))) float v8f;

namespace {
constexpr int kB  = 1048576;  // batch
constexpr int kD  = 17;       // input dim
constexpr int kH  = 8;        // expert hidden
constexpr int kO  = 6;        // output dim
constexpr int kE  = 4;        // experts
constexpr int kG1 = 64;       // gating hidden 1
constexpr int kG2 = 32;       // gating hidden 2
constexpr int kKP = 20;       // layer-1 K padded to multiple of 4

constexpr int kWaves       = 8;
constexpr int kThreads     = kWaves * 32;
constexpr int kRowsPerWave = 32;                     // two 16-sample WMMA tiles
constexpr int kRowsPerBlk  = kWaves * kRowsPerWave;  // 256
constexpr int kBlocks      = 1024;                   // persistent blocks
constexpr int kChunksPerBlk = kB / (kRowsPerBlk * kBlocks);  // 4 (exact, no tail)
constexpr int kH1S = 68;  // h1 row stride: mult of 4 (b128 stores), 68%64=4 -> conflict-free b64 reads
constexpr int kH2S = 36;  // h2 row stride: mult of 4 (b128 loads in tail)
constexpr int kW2S = 68;  // W2^T row stride: conflict-free b64 fragment reads
}  // namespace

__device__ __forceinline__ v8f wmma_f32_16x16x4(v2f a, v2f b, v8f c) {
    // (neg_a, A, neg_b, B, c_mod, C, reuse_a, reuse_b)
    return __builtin_amdgcn_wmma_f32_16x16x4_f32(false, a, false, b, (short)0, c,
                                                 false, false);
}

__global__ __launch_bounds__(kThreads)
void moe_gate_expert_kernel(const float* __restrict__ x,
                            const float* __restrict__ eW1, const float* __restrict__ eb1,
                            const float* __restrict__ eW2, const float* __restrict__ eb2,
                            const float* __restrict__ eW3, const float* __restrict__ eb3,
                            const float* __restrict__ gW1, const float* __restrict__ gb1,
                            const float* __restrict__ gW2, const float* __restrict__ gb2,
                            const float* __restrict__ gW3, const float* __restrict__ gb3,
                            float* __restrict__ out_pred,    // [B,6]
                            float* __restrict__ out_logits,  // [B,4]
                            const int chunksPerBlk)          // runtime: defeats full unroll
{
    // Transposed weights so A-fragments (K-pairs) are contiguous -> ds_load_b64
    __shared__ __attribute__((aligned(16))) float s_W1t[kG1][kKP];  // W1^T, K-pad rows 17..19 = 0
    __shared__ __attribute__((aligned(16))) float s_gb1[kG1];
    __shared__ __attribute__((aligned(16))) float s_W2t[kG2][kW2S]; // W2^T, cols 0..63 valid
    __shared__ __attribute__((aligned(16))) float s_gb2[kG2];
    __shared__ __attribute__((aligned(16))) float s_gW3[kG2][kE];   // 16B rows
    __shared__ __attribute__((aligned(16))) float s_gb3[kE];
    __shared__ __attribute__((aligned(16))) float s_eW1[kE][kD][kH];
    __shared__ __attribute__((aligned(16))) float s_eb1[kE][kH];
    __shared__ __attribute__((aligned(16))) float s_eW2[kE][kH][kH];
    __shared__ __attribute__((aligned(16))) float s_eb2[kE][kH];
    __shared__ __attribute__((aligned(16))) float s_eW3p[kE][kH][kH]; // cols 6,7 = 0
    __shared__ __attribute__((aligned(16))) float s_eb3p[kE][kH];     // cols 6,7 = 0
    __shared__ __attribute__((aligned(16))) float s_h1[kWaves][kRowsPerWave][kH1S]; // [sample][feat]
    __shared__ __attribute__((aligned(16))) float s_h2[kWaves][kRowsPerWave][kH2S]; // [sample][feat]

    const int tid = threadIdx.x;

    // ------------- stage (transposed) weights into LDS, once per block -------------
    for (int i = tid; i < kG1 * kKP; i += kThreads) {
        const int c = i / kKP, r = i - c * kKP;           // c = output feature, r = input k
        s_W1t[c][r] = (r < kD) ? gW1[r * kG1 + c] : 0.0f;
    }
    for (int i = tid; i < kG1; i += kThreads) s_gb1[i] = gb1[i];
    for (int i = tid; i < kG2 * kG1; i += kThreads) {
        const int m = i >> 6, k = i & 63;                 // m = output feature, k = input
        s_W2t[m][k] = gW2[k * kG2 + m];
    }
    for (int i = tid; i < kG2; i += kThreads) s_gb2[i] = gb2[i];
    for (int i = tid; i < kG2 * kE; i += kThreads) (&s_gW3[0][0])[i] = gW3[i];
    for (int i = tid; i < kE; i += kThreads) s_gb3[i] = gb3[i];
    for (int i = tid; i < kE * kD * kH; i += kThreads) (&s_eW1[0][0][0])[i] = eW1[i];
    for (int i = tid; i < kE * kH; i += kThreads) (&s_eb1[0][0])[i] = eb1[i];
    for (int i = tid; i < kE * kH * kH; i += kThreads) (&s_eW2[0][0][0])[i] = eW2[i];
    for (int i = tid; i < kE * kH; i += kThreads) (&s_eb2[0][0])[i] = eb2[i];
    for (int i = tid; i < kE * kH * kH; i += kThreads) {  // padded eW3: [e][k][8]
        const int e = i >> 6, rem = i & 63, k = rem >> 3, o = rem & 7;
        s_eW3p[e][k][o] = (o < kO) ? eW3[(e * kH + k) * kO + o] : 0.0f;
    }
    for (int i = tid; i < kE * kH; i += kThreads) {       // padded eb3
        const int e = i >> 3, o = i & 7;
        s_eb3p[e][o] = (o < kO) ? eb3[e * kO + o] : 0.0f;
    }
    __syncthreads();

    const int wave    = tid >> 5;
    const int lane    = tid & 31;
    const int lane_lo = lane & 15;
    const int half    = lane >> 4;

    float* __restrict__ h1w = &s_h1[wave][0][0];
    float* __restrict__ h2w = &s_h2[wave][0][0];

    // Persistent loop over chunks of 256 samples. Trip count is a RUNTIME value
    // so the compiler cannot fully unroll / cross-chunk pipeline the body (the
    // round-3/4 builds blew past 256 VGPRs and flooded the stream with
    // s_set_vgpr_msb). One body copy keeps <256 VGPRs and full occupancy.
#pragma clang loop unroll(disable)
    for (int ic = 0; ic < chunksPerBlk; ++ic) {
        const int rowBase = (blockIdx.x * chunksPerBlk + ic) * kRowsPerBlk
                          + wave * kRowsPerWave;

        // ==== layer 1 (transposed): h1^T[64x16] = W1^T[64x20] @ x^T[20x16], x2 tiles ====
        v8f acc1[2][4];  // [sample-tile][m-tile]
#pragma unroll
        for (int t = 0; t < 2; ++t)
#pragma unroll
            for (int mt = 0; mt < 4; ++mt)
#pragma unroll
                for (int j = 0; j < 8; ++j) acc1[t][mt][j] = 0.0f;

        const int s0 = rowBase + lane_lo;   // B-tile 0: this lane's sample (N index)
        const int s1 = s0 + 16;             // B-tile 1

#pragma unroll
        for (int kk = 0; kk < 4; ++kk) {    // c0 <= 15: always in-bounds -> b64-merged loads
            const int c0 = kk * 4 + half * 2;
            v2f b0, b1;
            b0.x = x[s0 * kD + c0    ];
            b0.y = x[s0 * kD + c0 + 1];
            b1.x = x[s1 * kD + c0    ];
            b1.y = x[s1 * kD + c0 + 1];
#pragma unroll
            for (int mt = 0; mt < 4; ++mt) {
                const v2f a = *(const v2f*)&s_W1t[mt * 16 + lane_lo][c0];  // ds_load_b64
                acc1[0][mt] = wmma_f32_16x16x4(a, b0, acc1[0][mt]);
                acc1[1][mt] = wmma_f32_16x16x4(a, b1, acc1[1][mt]);
            }
        }
        {   // K-tail kk=4: c0 = 16+2*half; only half==0 lane's .x is real (branchless select)
            const int c0 = 16 + half * 2;
            const float x0 = x[s0 * kD + 16];  // always-valid address, all lanes
            const float x1 = x[s1 * kD + 16];
            v2f b0, b1;
            b0.x = half ? 0.0f : x0;  b0.y = 0.0f;
            b1.x = half ? 0.0f : x1;  b1.y = 0.0f;
#pragma unroll
            for (int mt = 0; mt < 4; ++mt) {
                const v2f a = *(const v2f*)&s_W1t[mt * 16 + lane_lo][c0];  // rows 17..19 are 0
                acc1[0][mt] = wmma_f32_16x16x4(a, b0, acc1[0][mt]);
                acc1[1][mt] = wmma_f32_16x16x4(a, b1, acc1[1][mt]);
            }
        }

        // bias + ReLU; spill h1^T C-layout -> row-major h1[sample][feat], 2x ds_store_b128
#pragma unroll
        for (int t = 0; t < 2; ++t) {
            const int sloc = t * 16 + lane_lo;
#pragma unroll
            for (int mt = 0; mt < 4; ++mt) {
                const float* bp = &s_gb1[mt * 16 + half * 8];  // VGPR j <-> feature 8*half + j
                const v4f blo = *(const v4f*)bp;
                const v4f bhi = *(const v4f*)(bp + 4);
                v4f lo, hi;
#pragma unroll
                for (int j = 0; j < 4; ++j) {
                    float vl = acc1[t][mt][j]     + blo[j];
                    float vh = acc1[t][mt][4 + j] + bhi[j];
                    lo[j] = vl > 0.0f ? vl : 0.0f;
                    hi[j] = vh > 0.0f ? vh : 0.0f;
                }
                float* hp = &h1w[sloc * kH1S + mt * 16 + half * 8];
                *(v4f*)hp       = lo;
                *(v4f*)(hp + 4) = hi;
            }
        }

        // ==== layer 2 (transposed): h2^T[32x16] = W2^T[32x64] @ h1^T[64x16], x2 tiles ====
        v8f acc2[2][2];
#pragma unroll
        for (int t = 0; t < 2; ++t)
#pragma unroll
            for (int mt = 0; mt < 2; ++mt)
#pragma unroll
                for (int j = 0; j < 8; ++j) acc2[t][mt][j] = 0.0f;

#pragma unroll
        for (int kk = 0; kk < kG1 / 4; ++kk) {
            const int c0 = kk * 4 + half * 2;
            const v2f b0 = *(const v2f*)&h1w[lane_lo * kH1S + c0];
            const v2f b1 = *(const v2f*)&h1w[(16 + lane_lo) * kH1S + c0];
#pragma unroll
            for (int mt = 0; mt < 2; ++mt) {
                const v2f a = *(const v2f*)&s_W2t[mt * 16 + lane_lo][c0];
                acc2[0][mt] = wmma_f32_16x16x4(a, b0, acc2[0][mt]);
                acc2[1][mt] = wmma_f32_16x16x4(a, b1, acc2[1][mt]);
            }
        }

        // bias + ReLU; spill h2[sample][feat]
#pragma unroll
        for (int t = 0; t < 2; ++t) {
            const int sloc = t * 16 + lane_lo;
#pragma unroll
            for (int mt = 0; mt < 2; ++mt) {
                const float* bp = &s_gb2[mt * 16 + half * 8];
                const v4f blo = *(const v4f*)bp;
                const v4f bhi = *(const v4f*)(bp + 4);
                v4f lo, hi;
#pragma unroll
                for (int j = 0; j < 4; ++j) {
                    float vl = acc2[t][mt][j]     + blo[j];
                    float vh = acc2[t][mt][4 + j] + bhi[j];
                    lo[j] = vl > 0.0f ? vl : 0.0f;
                    hi[j] = vh > 0.0f ? vh : 0.0f;
                }
                float* hp = &h2w[sloc * kH2S + mt * 16 + half * 8];
                *(v4f*)hp       = lo;
                *(v4f*)(hp + 4) = hi;
            }
        }

        // ===== per-sample tail: layer 3 (32->4), argmax, selected expert =====
        const int srow = rowBase + lane;  // one sample per lane, 32 per wave
        const float* hrow = &h2w[lane * kH2S];

        v4f hq[8];
#pragma unroll
        for (int j = 0; j < 8; ++j) hq[j] = *(const v4f*)(hrow + j * 4);  // ds_load_b128

        v4f lacc = *(const v4f*)s_gb3;
#pragma unroll
        for (int j = 0; j < 8; ++j)
#pragma unroll
            for (int c = 0; c < 4; ++c) {
                const v4f w = *(const v4f*)&s_gW3[j * 4 + c][0];  // b128 (broadcast)
                lacc = lacc + w * hq[j][c];
            }

        int sel = 0;
        float best = lacc[0];
#pragma unroll
        for (int e = 1; e < kE; ++e) {  // strict '>' keeps first occurrence (argmax)
            if (lacc[e] > best) { best = lacc[e]; sel = e; }
        }
        *(v4f*)&out_logits[srow * kE] = lacc;  // coalesced global_store_b128

        float xin[kD];
#pragma unroll
        for (int c = 0; c < kD; ++c) xin[c] = x[srow * kD + c];

        // expert layer 1: 17 -> 8
        v4f aLo = *(const v4f*)&s_eb1[sel][0];
        v4f aHi = *(const v4f*)&s_eb1[sel][4];
#pragma unroll
        for (int c = 0; c < kD; ++c) {
            const float* wp = &s_eW1[sel][c][0];
            aLo = aLo + *(const v4f*)wp       * xin[c];
            aHi = aHi + *(const v4f*)(wp + 4) * xin[c];
        }
#pragma unroll
        for (int j = 0; j < 4; ++j) {
            aLo[j] = aLo[j] > 0.0f ? aLo[j] : 0.0f;
            aHi[j] = aHi[j] > 0.0f ? aHi[j] : 0.0f;
        }
        // expert layer 2: 8 -> 8
        v4f bLo = *(const v4f*)&s_eb2[sel][0];
        v4f bHi = *(const v4f*)&s_eb2[sel][4];
#pragma unroll
        for (int k = 0; k < kH; ++k) {
            const float hk = (k < 4) ? aLo[k] : aHi[k - 4];
            const float* wp = &s_eW2[sel][k][0];
            bLo = bLo + *(const v4f*)wp       * hk;
            bHi = bHi + *(const v4f*)(wp + 4) * hk;
        }
#pragma unroll
        for (int j = 0; j < 4; ++j) {
            bLo[j] = bLo[j] > 0.0f ? bLo[j] : 0.0f;
            bHi[j] = bHi[j] > 0.0f ? bHi[j] : 0.0f;
        }
        // expert layer 3: 8 -> 6 (zero-padded to 8)
        v4f oLo = *(const v4f*)&s_eb3p[sel][0];
        v4f oHi = *(const v4f*)&s_eb3p[sel][4];
#pragma unroll
        for (int k = 0; k < kH; ++k) {
            const float hk = (k < 4) ? bLo[k] : bHi[k - 4];
            const float* wp = &s_eW3p[sel][k][0];
            oLo = oLo + *(const v4f*)wp       * hk;
            oHi = oHi + *(const v4f*)(wp + 4) * hk;
        }
        float* op = &out_pred[srow * kO];  // srow*6 even -> 8B aligned v2f stores
        v2f p0; p0.x = oLo[0]; p0.y = oLo[1];
        v2f p1; p1.x = oLo[2]; p1.y = oLo[3];
        v2f p2; p2.x = oHi[0]; p2.y = oHi[1];
        *(v2f*)op       = p0;
        *(v2f*)(op + 2) = p1;
        *(v2f*)(op + 4) = p2;
    }
}

extern "C" void kernel_launch(void* const* d_in, const int* in_sizes, int n_in,
                              void* d_out, int out_size, void* d_ws, size_t ws_size,
                              hipStream_t stream) {
    const float* x   = (const float*)d_in[0];
    const float* eW1 = (const float*)d_in[1];
    const float* eb1 = (const float*)d_in[2];
    const float* eW2 = (const float*)d_in[3];
    const float* eb2 = (const float*)d_in[4];
    const float* eW3 = (const float*)d_in[5];
    const float* eb3 = (const float*)d_in[6];
    const float* gW1 = (const float*)d_in[7];
    const float* gb1 = (const float*)d_in[8];
    const float* gW2 = (const float*)d_in[9];
    const float* gb2 = (const float*)d_in[10];
    const float* gW3 = (const float*)d_in[11];
    const float* gb3 = (const float*)d_in[12];

    float* out_pred   = (float*)d_out;                    // [B,6] first
    float* out_logits = (float*)d_out + (size_t)kB * kO;  // then [B,4]

    dim3 grid(kBlocks);  // persistent: 1024 blocks x 4 chunks x 256 rows = 2^20, no tail
    moe_gate_expert_kernel<<<grid, kThreads, 0, stream>>>(
        x, eW1, eb1, eW2, eb2, eW3, eb3,
        gW1, gb1, gW2, gb2, gW3, gb3,
        out_pred, out_logits, kChunksPerBlk);
}